// ProFITi_30090540876050
// MI455X (gfx1250) — compile-verified
//
#include <hip/hip_runtime.h>
#include <math.h>

typedef float v2f __attribute__((ext_vector_type(2)));
typedef float v8f __attribute__((ext_vector_type(8)));
typedef unsigned int u32x4 __attribute__((ext_vector_type(4)));
typedef int i32x4 __attribute__((ext_vector_type(4)));
typedef int i32x8 __attribute__((ext_vector_type(8)));

#define S_LEN 2048
#define BATCH 8
#define DIM   128
#define NROW  (BATCH * S_LEN)   // 16384
#define NLAY  3

#if __has_builtin(__builtin_amdgcn_tensor_load_to_lds) && \
    __has_builtin(__builtin_amdgcn_s_wait_tensorcnt)
#define USE_TDM 1
#endif

// ---------------------------------------------------------------------------
// Tensor Data Mover: issue a 2D f32 tile DMA global->LDS described by a D#.
// Group0: count=1 | lds_addr | global_addr[56:0] | type=2
// Group1: data_size=4B, pad cfg, tensor_dim0/1, tile_dim0/1, dim0 stride
// LDS padding is done by the TDM itself (pad_interval/pad_amount), matching
// the +1-DWORD-per-row bank-conflict padding of the LDS arrays.
// ---------------------------------------------------------------------------
__device__ __forceinline__ void tdm_load_2d_f32(
        const void* gptr, unsigned lds_off_bytes,
        unsigned tile_w, unsigned tile_h, unsigned row_stride_elems,
        unsigned pad_interval_code, unsigned pad_amount_code)
{
#ifdef USE_TDM
    unsigned long long ga = (unsigned long long)(uintptr_t)gptr;
    u32x4 g0;
    g0.x = 0x1u;                                     // count=1 (user D#)
    g0.y = lds_off_bytes;                            // D#.lds_addr (bytes)
    g0.z = (unsigned)(ga & 0xffffffffu);             // global_addr[31:0]
    g0.w = (unsigned)((ga >> 32) & 0x01ffffffu)      // global_addr[56:32]
         | (2u << 30);                               // type = 2 ("image")
    i32x8 g1;
    unsigned d0 = (2u << 16)                         // data_size = 4 bytes
                | (1u << 20)                         // pad_enable
                | (pad_interval_code << 22)
                | (pad_amount_code << 25);
    g1[0] = (int)d0;
    g1[1] = (int)((tile_w & 0xffffu) << 16);         // tensor_dim0[15:0]
    g1[2] = (int)(((tile_w >> 16) & 0xffffu)         // tensor_dim0[31:16]
                | ((tile_h & 0xffffu) << 16));       // tensor_dim1[15:0]
    g1[3] = (int)(((tile_h >> 16) & 0xffffu)         // tensor_dim1[31:16]
                | ((tile_w & 0xffffu) << 16));       // tile_dim0
    g1[4] = (int)(tile_h & 0xffffu);                 // tile_dim1 (tile_dim2=0)
    g1[5] = (int)row_stride_elems;                   // tensor_dim0_stride[31:0]
    g1[6] = 0;
    g1[7] = 0;
    i32x4 z4 = {0, 0, 0, 0};
#if __clang_major__ >= 23
    i32x8 z8 = {0, 0, 0, 0, 0, 0, 0, 0};
    __builtin_amdgcn_tensor_load_to_lds(g0, g1, z4, z4, z8, 0);
#else
    __builtin_amdgcn_tensor_load_to_lds(g0, g1, z4, z4, 0);
#endif
#else
    (void)gptr; (void)lds_off_bytes; (void)tile_w; (void)tile_h;
    (void)row_stride_elems; (void)pad_interval_code; (void)pad_amount_code;
#endif
}

// ---------------------------------------------------------------------------
// GEMM: C[NROW x 128] = act(A[NROW x 128] @ W[128 x 128] + bias[128])
// f32 WMMA path (V_WMMA_F32_16X16X4_F32): per block, 128x128 C tile,
// 8 waves x (16 rows x 128 cols), K staged through LDS in 32-wide chunks
// via the Tensor Data Mover (fallback: cooperative vector loads).
// A-frag (16x4 f32, 2 VGPR): lanes 0-15 row=lane k={k0,k0+1}; lanes 16-31
// row=lane-16 k={k0+2,k0+3}. B-frag (4x16) is the dual layout.
// ---------------------------------------------------------------------------
__global__ __launch_bounds__(256) void gemm128_wmma(
        const float* __restrict__ A, const float* __restrict__ W,
        const float* __restrict__ bias, float* __restrict__ C, int relu)
{
    __shared__ float Hs[128][33];   // +1 pad: column reads hit distinct banks
    __shared__ float Ws[32][129];

    const int tid  = threadIdx.x;
    const int lane = tid & 31;
    const int wv   = tid >> 5;          // 8 waves
    const int rowBase = blockIdx.x * 128;
    const int l16  = lane & 15;
    const int koff = (lane >> 4) * 2;   // 0 or 2

    v8f acc[8];
    #pragma unroll
    for (int n = 0; n < 8; ++n) acc[n] = (v8f)0.0f;

    for (int kc = 0; kc < 128; kc += 32) {
#ifdef USE_TDM
        if (tid == 0) {
            // H tile: 128 rows x 32 f32, row stride 128; +1 DWORD pad / 32 DWORDs
            tdm_load_2d_f32(A + (size_t)rowBase * 128 + kc,
                            (unsigned)(uintptr_t)&Hs[0][0],
                            /*tile_w=*/32, /*tile_h=*/128, /*stride=*/128,
                            /*interval 32 dw=*/4, /*amount 1 dw=*/0);
            // W tile: 32 rows x 128 f32, row stride 128; +1 DWORD pad / 128 DWORDs
            tdm_load_2d_f32(W + (size_t)kc * 128,
                            (unsigned)(uintptr_t)&Ws[0][0],
                            /*tile_w=*/128, /*tile_h=*/32, /*stride=*/128,
                            /*interval 128 dw=*/6, /*amount 1 dw=*/0);
            __builtin_amdgcn_s_wait_tensorcnt(0);
        }
        __syncthreads();
#else
        #pragma unroll
        for (int i = 0; i < 16; ++i) {
            int idx = tid + i * 256;
            int r = idx >> 5, c = idx & 31;
            Hs[r][c] = A[(size_t)(rowBase + r) * 128 + kc + c];
        }
        #pragma unroll
        for (int i = 0; i < 16; ++i) {
            int idx = tid + i * 256;
            int r = idx >> 7, c = idx & 127;
            Ws[r][c] = W[(size_t)(kc + r) * 128 + c];
        }
        __syncthreads();
#endif

        #pragma unroll
        for (int k0 = 0; k0 < 32; k0 += 4) {
            v2f a;
            a.x = Hs[wv * 16 + l16][k0 + koff];
            a.y = Hs[wv * 16 + l16][k0 + koff + 1];
            #pragma unroll
            for (int n = 0; n < 8; ++n) {
                v2f b;
                b.x = Ws[k0 + koff][n * 16 + l16];
                b.y = Ws[k0 + koff + 1][n * 16 + l16];
                acc[n] = __builtin_amdgcn_wmma_f32_16x16x4_f32(
                    false, a, false, b, (short)0, acc[n], false, false);
            }
        }
        __syncthreads();   // WAR fence before next tile DMA / staging
    }

    // C layout: VGPR v -> lanes 0-15: M=v, lanes 16-31: M=v+8; N = lane&15
    const int halfsel = lane >> 4;
    #pragma unroll
    for (int n = 0; n < 8; ++n) {
        int col = n * 16 + l16;
        float bv = bias[col];
        #pragma unroll
        for (int v = 0; v < 8; ++v) {
            int row = rowBase + wv * 16 + v + halfsel * 8;
            float val = acc[n][v] + bv;
            if (relu) val = fmaxf(val, 0.0f);
            C[(size_t)row * 128 + col] = val;
        }
    }
}

// ---------------------------------------------------------------------------
// out[row] = f( dot(X[row,:128], w) + *b ); mode 1 -> exp(tanh(x))
// One wave per row, 4 elements/lane + shuffle reduction.
// ---------------------------------------------------------------------------
__global__ __launch_bounds__(256) void rowdot128(
        const float* __restrict__ X, const float* __restrict__ w,
        const float* __restrict__ bscal, float* __restrict__ out, int mode)
{
    const int lane = threadIdx.x & 31;
    const int wv   = threadIdx.x >> 5;
    const int row  = blockIdx.x * 8 + wv;
    const float* xr = X + (size_t)row * 128;
    float s = xr[lane]      * w[lane]
            + xr[lane + 32] * w[lane + 32]
            + xr[lane + 64] * w[lane + 64]
            + xr[lane + 96] * w[lane + 96];
    #pragma unroll
    for (int off = 16; off; off >>= 1) s += __shfl_xor(s, off, 32);
    if (lane == 0) {
        float v = s + bscal[0];
        if (mode) v = expf(tanhf(v));
        out[row] = v;
    }
}

// Z = (Y*MQ - EL0)*MQ ; LDJ = 1
__global__ void init_z(const float* __restrict__ Y, const float* __restrict__ MQ,
                       const float* __restrict__ EL0, float* __restrict__ Z,
                       float* __restrict__ LDJ)
{
    int i = blockIdx.x * blockDim.x + threadIdx.x;
    if (i < NROW) {
        float mq = MQ[i];
        Z[i]   = (Y[i] * mq - EL0[i]) * mq;
        LDJ[i] = 1.0f;
    }
}

__device__ __forceinline__ float softplusf(float x) {
    if (x > 20.0f)  return x;
    if (x < -20.0f) return expf(x);
    return log1pf(expf(x));
}

// ---------------------------------------------------------------------------
// Sequential flow scan: one block per batch, 128 threads (thread j = channel j
// of the running prefix state P). Z lives in LDS for all 3 layers.
//   Z_new[s] = scale*q_s·P + (softplus(scale*q_s·k_s)+1e-3)*Z[s],  P += Z[s]*k_s
// followed by masking, affine (theta/phi), shiesh, and LDJ accumulation.
// ---------------------------------------------------------------------------
__global__ __launch_bounds__(128) void flow_scan(
        const float* __restrict__ Q, const float* __restrict__ K,
        const float* __restrict__ TH, const float* __restrict__ PH,
        const float* __restrict__ MQ, float* __restrict__ Z,
        float* __restrict__ LDJ)
{
    __shared__ float Zsh[S_LEN];
    __shared__ float red[8];

    const int b    = blockIdx.x;
    const int tid  = threadIdx.x;
    const int lane = tid & 31;
    const int wv   = tid >> 5;
    const float scale = 0.08838834764831845f;  // 128^-0.5
    const float e1    = 2.7182818284590452f;

    for (int i = tid; i < S_LEN; i += 128) Zsh[i] = Z[b * S_LEN + i];
    __syncthreads();

    for (int l = 0; l < NLAY; ++l) {
        const float* Ql  = Q  + ((size_t)l * NROW + (size_t)b * S_LEN) * 128;
        const float* Kl  = K  + ((size_t)l * NROW + (size_t)b * S_LEN) * 128;
        const float* thl = TH + (size_t)l * NROW + (size_t)b * S_LEN;
        const float* phl = PH + (size_t)l * NROW + (size_t)b * S_LEN;

        float P = 0.0f;
        for (int s = 0; s < S_LEN; ++s) {
            float qv   = Ql[(size_t)s * 128 + tid];
            float kv   = Kl[(size_t)s * 128 + tid];
            float zold = Zsh[s];
            if (s + 8 < S_LEN) {   // global_prefetch_b8 for upcoming rows
                __builtin_prefetch(&Ql[(size_t)(s + 8) * 128 + tid], 0, 0);
                __builtin_prefetch(&Kl[(size_t)(s + 8) * 128 + tid], 0, 0);
            }
            float dp = qv * P, dk = qv * kv;
            #pragma unroll
            for (int off = 16; off; off >>= 1) {
                dp += __shfl_xor(dp, off, 32);
                dk += __shfl_xor(dk, off, 32);
            }
            if (lane == 0) { red[wv] = dp; red[4 + wv] = dk; }
            __syncthreads();   // also orders Zsh[s] reads before tid0's write

            if (tid == 0) {
                float qdotP = red[0] + red[1] + red[2] + red[3];
                float qdotK = red[4] + red[5] + red[6] + red[7];
                float diag  = softplusf(scale * qdotK) + 1e-3f;
                float mq    = MQ[b * S_LEN + s];
                float z     = (scale * qdotP + diag * zold) * mq;
                float ldj   = LDJ[b * S_LEN + s] + logf(diag) * mq;
                float t = thl[s], p = phl[s];
                z   = z * t + p;
                ldj += t * mq;
                if (fabsf(z) > 5.0f) {
                    z = z + (z > 0.0f ? 1.0f : -1.0f);   // shiesh linear tail
                } else {
                    float sh = sinhf(z), ch = coshf(z);
                    float u  = 1.0f + e1 * sh;
                    ldj += (logf(e1 * ch) - logf(u * u)) * mq;
                    z = asinhf(e1 * sh);
                }
                Zsh[s]             = z;
                LDJ[b * S_LEN + s] = ldj;
            }
            P += zold * kv;    // prefix state uses pre-update Z
            __syncthreads();
        }
    }
    for (int i = tid; i < S_LEN; i += 128) Z[b * S_LEN + i] = Zsh[i];
}

// ---------------------------------------------------------------------------
extern "C" void kernel_launch(void* const* d_in, const int* in_sizes, int n_in,
                              void* d_out, int out_size, void* d_ws, size_t ws_size,
                              hipStream_t stream)
{
    (void)in_sizes; (void)n_in; (void)out_size; (void)ws_size;

    const float* H   = (const float*)d_in[0];
    const float* Y   = (const float*)d_in[1];
    const float* MQ  = (const float*)d_in[2];
    const float* Wq  = (const float*)d_in[3];
    const float* bq  = (const float*)d_in[4];
    const float* Wk  = (const float*)d_in[5];
    const float* bk  = (const float*)d_in[6];
    const float* tW1 = (const float*)d_in[7];
    const float* tb1 = (const float*)d_in[8];
    const float* tW2 = (const float*)d_in[9];
    const float* tb2 = (const float*)d_in[10];
    const float* pW1 = (const float*)d_in[11];
    const float* pb1 = (const float*)d_in[12];
    const float* pW2 = (const float*)d_in[13];
    const float* pb2 = (const float*)d_in[14];
    const float* wi  = (const float*)d_in[15];
    const float* bi  = (const float*)d_in[16];

    float* Zout = (float*)d_out;           // [B,S]
    float* LDJ  = Zout + NROW;             // [B,S]

    // workspace carve-up (floats): Q,K persist across scan; T reused per MLP
    float* ws    = (float*)d_ws;
    float* Qbuf  = ws;                                   // 3*16384*128
    float* Kbuf  = Qbuf  + (size_t)NLAY * NROW * DIM;    // 3*16384*128
    float* Tbuf  = Kbuf  + (size_t)NLAY * NROW * DIM;    // 16384*128 scratch
    float* theta = Tbuf  + (size_t)NROW * DIM;           // 3*16384
    float* phi   = theta + (size_t)NLAY * NROW;          // 3*16384
    float* el0   = phi   + (size_t)NLAY * NROW;          // 16384

    const dim3 gGemm(128), bGemm(256);
    const dim3 gDot(NROW / 8), bDot(256);

    // EL0 = H @ wi + bi ; Z/LDJ init
    rowdot128<<<gDot, bDot, 0, stream>>>(H, wi, bi, el0, 0);
    init_z<<<NROW / 256, 256, 0, stream>>>(Y, MQ, el0, Zout, LDJ);

    for (int l = 0; l < NLAY; ++l) {
        const size_t wo = (size_t)l * DIM * DIM, bo = (size_t)l * DIM;
        float* Ql = Qbuf + (size_t)l * NROW * DIM;
        float* Kl = Kbuf + (size_t)l * NROW * DIM;
        gemm128_wmma<<<gGemm, bGemm, 0, stream>>>(H, Wq + wo, bq + bo, Ql, 0);
        gemm128_wmma<<<gGemm, bGemm, 0, stream>>>(H, Wk + wo, bk + bo, Kl, 0);
        // theta = exp(tanh(relu(H@tW1+tb1) @ tW2 + tb2))
        gemm128_wmma<<<gGemm, bGemm, 0, stream>>>(H, tW1 + wo, tb1 + bo, Tbuf, 1);
        rowdot128<<<gDot, bDot, 0, stream>>>(Tbuf, tW2 + bo, tb2 + l,
                                             theta + (size_t)l * NROW, 1);
        // phi = relu(H@pW1+pb1) @ pW2 + pb2
        gemm128_wmma<<<gGemm, bGemm, 0, stream>>>(H, pW1 + wo, pb1 + bo, Tbuf, 1);
        rowdot128<<<gDot, bDot, 0, stream>>>(Tbuf, pW2 + bo, pb2 + l,
                                             phi + (size_t)l * NROW, 0);
    }

    flow_scan<<<BATCH, 128, 0, stream>>>(Qbuf, Kbuf, theta, phi, MQ, Zout, LDJ);
}